// SimilarityMatrix_10368051052605
// MI455X (gfx1250) — compile-verified
//
#include <hip/hip_runtime.h>

typedef __attribute__((ext_vector_type(2))) float v2f;
typedef __attribute__((ext_vector_type(4))) float v4f;
typedef __attribute__((ext_vector_type(8))) float v8f;
typedef __attribute__((ext_vector_type(4))) int   v4i;

#define NB    64      // batches
#define TC    512     // context length
#define TQ    64      // question length
#define DD    512     // feature dim
#define QPAD  516     // padded LDS row stride (floats): bank stride 4 -> conflict-free
#define MROWS 128     // context rows per workgroup
#define NGRP  (MROWS / 16)                    // 8 accumulator tiles per wave
#define LDS_Q_FLOATS (TQ * QPAD)              // 33024
#define LDS_W_OFF    LDS_Q_FLOATS
#define LDS_FLOATS   (LDS_Q_FLOATS + 3 * DD)  // 138240 bytes

#if __has_builtin(__builtin_amdgcn_global_load_async_to_lds_b128)
#define HAVE_ASYNC_LDS 1
#else
#define HAVE_ASYNC_LDS 0
#endif

typedef __attribute__((address_space(1))) v4i* gptr_v4i;
typedef __attribute__((address_space(3))) v4i* lptr_v4i;

__global__ __launch_bounds__(128, 2)
void simmat_wmma_f32(const float* __restrict__ ctx,
                     const float* __restrict__ qst,
                     const float* __restrict__ ker,
                     float* __restrict__ out)
{
    extern __shared__ float lds[];
    const int tid  = threadIdx.x;
    const int lane = tid & 31;
    const int wv   = tid >> 5;            // wave id 0..3 -> Tq 16-col block
    const int wg   = blockIdx.x;
    const int b    = wg >> 2;             // batch
    const int row0 = (wg & 3) * MROWS;    // 128-row context block

    // ---- stage question tile (64 x 512 f32) + kernel vector (1536 f32) in LDS ----
    const float* qb = qst + (size_t)b * TQ * DD;
#if HAVE_ASYNC_LDS
    for (int idx = tid; idx < TQ * (DD / 4); idx += 128) {
        int r = idx >> 7;                 // 128 float4 per row
        int c = idx & 127;
        __builtin_amdgcn_global_load_async_to_lds_b128(
            (gptr_v4i)(qb + r * DD + c * 4),
            (lptr_v4i)(lds + r * QPAD + c * 4),
            0, 0);
    }
#else
    for (int idx = tid; idx < TQ * (DD / 4); idx += 128) {
        int r = idx >> 7;
        int c = idx & 127;
        v4f v = *(const v4f*)(qb + r * DD + c * 4);
        *(v4f*)(lds + r * QPAD + c * 4) = v;
    }
#endif
    for (int idx = tid; idx < (3 * DD) / 4; idx += 128) {
        v4f v = *(const v4f*)(ker + idx * 4);
        *(v4f*)(lds + LDS_W_OFF + idx * 4) = v;
    }
#if HAVE_ASYNC_LDS
#if __has_builtin(__builtin_amdgcn_s_wait_asynccnt)
    __builtin_amdgcn_s_wait_asynccnt(0);
#else
    asm volatile("s_wait_asynccnt 0x0" ::: "memory");
#endif
#endif
    __syncthreads();

    const int ln = lane & 15;             // row/col within 16-tile
    const int hi = lane >> 4;             // half-wave -> k offset +2
    const int n0 = wv * 16;

    const float* wc   = lds + LDS_W_OFF;
    const float* wq   = lds + LDS_W_OFF + DD;
    const float* wm   = lds + LDS_W_OFF + 2 * DD;
    const float* qrow = lds + (n0 + ln) * QPAD;

    // A-fragment rows: 8 groups of 16 rows; lane ln reads row (row0 + 16g + ln)
    const float* arow = ctx + ((size_t)b * TC + row0 + ln) * DD;

    v8f   acc[NGRP];
    float sc[NGRP];
    #pragma unroll
    for (int g = 0; g < NGRP; ++g) { acc[g] = (v8f){}; sc[g] = 0.f; }
    float sq = 0.f;

    #pragma unroll 2
    for (int k0 = 0; k0 < DD; k0 += 4) {
        const int koff = k0 + 2 * hi;

        v2f wm2 = *(const v2f*)(wm + koff);
        v2f wc2 = *(const v2f*)(wc + koff);
        v2f wq2 = *(const v2f*)(wq + koff);

        v2f b2 = *(const v2f*)(qrow + koff);   // B frag: Q[n][k], Q[n][k+1]
        v2f bs = b2 * wm2;                     // fold w_m into B
        sq += b2.x * wq2.x + b2.y * wq2.y;     // s_q partial

        v2f a[NGRP];
        #pragma unroll
        for (int g = 0; g < NGRP; ++g)
            a[g] = *(const v2f*)(arow + g * 16 * DD + koff);

        #pragma unroll
        for (int g = 0; g < NGRP; ++g)
            sc[g] += a[g].x * wc2.x + a[g].y * wc2.y;   // s_c partials

        // D = A * (w_m .* Q^T) + D   (exact f32 path), 8 WMMA per B fragment
        #pragma unroll
        for (int g = 0; g < NGRP; ++g)
            acc[g] = __builtin_amdgcn_wmma_f32_16x16x4_f32(
                false, a[g], false, bs, (short)0, acc[g], false, false);
    }

    // combine the two half-wave K partials (wave32: pairs (l, l^16))
    sq += __shfl_xor(sq, 16);
    #pragma unroll
    for (int g = 0; g < NGRP; ++g) sc[g] += __shfl_xor(sc[g], 16);
    // lane l now holds: sq -> s_q[n0 + (l&15)], sc[g] -> s_c[row0 + 16g + (l&15)]

    // C/D 16x16 f32 layout: VGPR v -> row v + 8*hi; col = lane&15
    #pragma unroll
    for (int g = 0; g < NGRP; ++g) {
        #pragma unroll
        for (int v = 0; v < 8; ++v) {
            float scv = __shfl(sc[g], v + 8 * hi);
            int   row = row0 + 16 * g + v + 8 * hi;
            out[((size_t)b * TC + row) * TQ + n0 + ln] = acc[g][v] + scv + sq;
        }
    }
}

extern "C" void kernel_launch(void* const* d_in, const int* in_sizes, int n_in,
                              void* d_out, int out_size, void* d_ws, size_t ws_size,
                              hipStream_t stream) {
    (void)in_sizes; (void)n_in; (void)d_ws; (void)ws_size; (void)out_size;
    const float* ctx = (const float*)d_in[0];   // (64, 512, 512) f32
    const float* qst = (const float*)d_in[1];   // (64, 64, 512)  f32
    const float* ker = (const float*)d_in[2];   // (1536, 1)      f32
    float* out = (float*)d_out;                 // (64, 512, 64)  f32

    dim3 grid(NB * (TC / MROWS));               // 256 workgroups
    dim3 block(128);                            // 4 waves (wave32)
    size_t ldsBytes = (size_t)LDS_FLOATS * sizeof(float);
    simmat_wmma_f32<<<grid, block, ldsBytes, stream>>>(ctx, qst, ker, out);
}